// ClassConditionalBatchNorm2d_76192719831881
// MI455X (gfx1250) — compile-verified
//
#include <hip/hip_runtime.h>

// Class-conditional BatchNorm2d, eval path with alpha=0.5.
// B=128, C=128, H=W=56 (HW=3136), NUM_CLASSES=100 in the reference; sizes are
// derived from in_sizes so the kernel stays general.
//
// Memory-bound streaming kernel: 411 MB of traffic at 23.3 TB/s -> ~17.6 us
// floor. One block per (b,c) plane via a 2D grid (no integer division in the
// preamble), block-uniform scalar stat math (lowered to s_load + v_rsq), then
// a b128 non-temporal load -> FMA -> b128 non-temporal store stream with a
// gfx1250 global_prefetch for the next stride.

#define CCBN_EPS   1e-5f
#define CCBN_ALPHA 0.5f
#define CCBN_BLOCK 256

typedef __attribute__((ext_vector_type(4))) float v4f;

__global__ __launch_bounds__(CCBN_BLOCK) void ccbn_eval_kernel(
    const float* __restrict__ x,
    const int*   __restrict__ labels,
    const float* __restrict__ weight,
    const float* __restrict__ bias,
    const float* __restrict__ gmean,
    const float* __restrict__ gvar,
    const float* __restrict__ cmean,
    const float* __restrict__ cvar,
    float*       __restrict__ out,
    int C, int HW)
{
    const int c = blockIdx.x;            // channel
    const int b = blockIdx.y;            // batch sample

    // Block-uniform stat math -> scalar loads + a handful of VALU ops.
    const int lbl = labels[b];
    const float m  = (1.0f - CCBN_ALPHA) * gmean[c] + CCBN_ALPHA * cmean[lbl * C + c];
    const float vv = (1.0f - CCBN_ALPHA) * gvar[c]  + CCBN_ALPHA * cvar[lbl * C + c];
    const float inv   = rsqrtf(vv + CCBN_EPS);
    const float scale = inv * weight[c];
    const float shift = fmaf(-m, scale, bias[c]);

    const size_t base = ((size_t)b * (size_t)C + (size_t)c) * (size_t)HW;

    if ((HW & 3) == 0) {
        // Vectorized streaming path: 16B per lane per iteration.
        const v4f* __restrict__ xv = (const v4f*)(x + base);
        v4f*       __restrict__ ov = (v4f*)(out + base);
        const int n4 = HW >> 2;
        for (int i = threadIdx.x; i < n4; i += CCBN_BLOCK) {
            // gfx1250 prefetch path (speculative; OOB addresses are dropped).
            __builtin_prefetch(&xv[i + CCBN_BLOCK], 0, 0);
            v4f val = __builtin_nontemporal_load(&xv[i]);
            v4f r;
            r.x = fmaf(val.x, scale, shift);
            r.y = fmaf(val.y, scale, shift);
            r.z = fmaf(val.z, scale, shift);
            r.w = fmaf(val.w, scale, shift);
            __builtin_nontemporal_store(r, &ov[i]);
        }
    } else {
        // Generic scalar fallback (not taken for the reference shapes).
        for (int i = threadIdx.x; i < HW; i += CCBN_BLOCK) {
            float val = __builtin_nontemporal_load(x + base + i);
            __builtin_nontemporal_store(fmaf(val, scale, shift), out + base + i);
        }
    }
}

extern "C" void kernel_launch(void* const* d_in, const int* in_sizes, int n_in,
                              void* d_out, int out_size, void* d_ws, size_t ws_size,
                              hipStream_t stream) {
    const float* x      = (const float*)d_in[0];
    const int*   labels = (const int*)  d_in[1];
    const float* weight = (const float*)d_in[2];
    const float* bias   = (const float*)d_in[3];
    const float* gmean  = (const float*)d_in[4];
    const float* gvar   = (const float*)d_in[5];
    const float* cmean  = (const float*)d_in[6];
    const float* cvar   = (const float*)d_in[7];
    float* out = (float*)d_out;

    const int B  = in_sizes[1];                 // labels: [B]
    const int C  = in_sizes[2];                 // weight: [C]
    const int HW = in_sizes[0] / (B * C);       // x: [B,C,H,W]

    dim3 grid(C, B);         // blockIdx.x = c, blockIdx.y = b -> no int division
    dim3 block(CCBN_BLOCK);  // 8 wave32 waves per block
    ccbn_eval_kernel<<<grid, block, 0, stream>>>(
        x, labels, weight, bias, gmean, gvar, cmean, cvar, out, C, HW);
}